// RegressionTransformerPyG_11845519802382
// MI455X (gfx1250) — compile-verified
//
#include <hip/hip_runtime.h>
#include <hip/hip_bf16.h>

// ---------------------------------------------------------------------------
// RegressionTransformerPyG on MI455X (gfx1250), wave32, WMMA fp32 path,
// W-panel staged into LDS via the Tensor Data Mover (TENSOR_LOAD_TO_LDS).
// N=20000 nodes, E=320000 edges, G=2000 graphs, H_DIM=64, HEADS=4, HC=256.
// ---------------------------------------------------------------------------

typedef __attribute__((ext_vector_type(2))) float v2f;
typedef __attribute__((ext_vector_type(8))) float v8f;
typedef __attribute__((ext_vector_type(4))) unsigned int v4u;
typedef __attribute__((ext_vector_type(4))) int v4i;
typedef __attribute__((ext_vector_type(8))) int v8i;

#define HC    256
#define HEADS 4
#define HD    64
#define KCHUNK 256   // K-chunk staged in LDS: panel = KCHUNK x 16 floats (16 KB)

#if __has_builtin(__builtin_amdgcn_tensor_load_to_lds) && \
    __has_builtin(__builtin_amdgcn_s_wait_tensorcnt)
#define USE_TDM 1
#if __has_include(<hip/amd_detail/amd_gfx1250_TDM.h>)
#define TDM_6ARG 1
#endif
#endif

// ---- monotonic float <-> uint encoding for atomic min/max on floats -------
__device__ __forceinline__ unsigned fenc(float f) {
  unsigned u = __float_as_uint(f);
  return (u & 0x80000000u) ? ~u : (u | 0x80000000u);
}
__device__ __forceinline__ float fdec(unsigned e) {
  unsigned u = (e & 0x80000000u) ? (e & 0x7FFFFFFFu) : ~e;
  return __uint_as_float(u);
}

// ---------------------------------------------------------------------------
// fp32 WMMA GEMM:  C[M x N] = A[M x K] @ W[K x N] + bias[N]
// Caller guarantees: K % 4 == 0, N % 16 == 0.
// Block = 128 threads (4 waves); block tile = 64(M) x 16(N); wave w owns rows
// [m0+16w, m0+16w+16). W[K x 16] panel staged in LDS per KCHUNK:
//   - TDM path: wave 0 issues TENSOR_LOAD_TO_LDS (2D tile kc x 16, row stride
//     N elements) and s_wait_tensorcnt 0; barrier publishes the panel.
//   - fallback: cooperative vector loads.
// LDS layout row-major Wp[k][16] (what TDM writes): B-frag = two ds_load_b32.
//
// V_WMMA_F32_16X16X4_F32 fragment layout (ISA 7.12.2):
//   A (16x4):  lane = (k>=2)*16 + m, vgpr = k&1  -> per-lane v2f A[m][2*khalf]
//   B (4x16):  per-lane v2f = W[2*khalf + {0,1}][col]
//   C/D:       lane holds col; vgpr r -> row r + 8*(lane>=16)
// ---------------------------------------------------------------------------
__global__ __launch_bounds__(128)
void gemm_wmma_f32(const float* __restrict__ A, const float* __restrict__ Wm,
                   const float* __restrict__ bias, float* __restrict__ C,
                   int M, int K, int N) {
  __shared__ float Wp[KCHUNK * 16];

  const int tid   = threadIdx.x;
  const int lane  = tid & 31;
  const int wv    = tid >> 5;
  const int idx   = lane & 15;   // A row within tile / B,C col within tile
  const int khalf = lane >> 4;   // 0 -> K{0,1}, 1 -> K{2,3}

  const int n0  = blockIdx.y * 16;
  const int m0  = blockIdx.x * 64 + wv * 16;
  const int row = m0 + idx;
  const int rowc = (row < M) ? row : (M - 1);           // branchless clamp
  const float* __restrict__ Arow = A + (size_t)rowc * K + 2 * khalf;
  const float* __restrict__ Bcol = &Wp[idx];
  const int col = n0 + idx;

  v8f acc = {};
  for (int kb = 0; kb < K; kb += KCHUNK) {
    const int kc = (K - kb < KCHUNK) ? (K - kb) : KCHUNK;

#if defined(USE_TDM)
    if (wv == 0) {
      // ---- Tensor DMA descriptor (ISA 8.3/8.4), 2D tile kc x 16 floats ----
      const unsigned long long ga =
          (unsigned long long)(size_t)(Wm + (size_t)kb * N + n0);
      const unsigned lds_addr = (unsigned)(size_t)(void*)Wp;  // low 32 = LDS off
      v4u g0;
      g0.x = 1u;                                   // count=1, user descriptor
      g0.y = lds_addr;                             // lds_addr [63:32]
      g0.z = (unsigned)(ga & 0xFFFFFFFFu);         // global_addr [95:64]
      g0.w = (unsigned)((ga >> 32) & 0x01FFFFFFu)  // global_addr [120:96]
             | 0x80000000u;                        // type=2 ("image") [127:126]
      const unsigned td0 = (unsigned)N;            // tensor_dim0 (elements)
      const unsigned td1 = (unsigned)kc;           // tensor_dim1
      const unsigned long long st0 = (unsigned long long)N;  // dim0 stride
      v8i g1;
      g1[0] = (int)(2u << 16);                     // data_size = 4 bytes
      g1[1] = (int)((td0 & 0xFFFFu) << 16);        // tensor_dim0 lo
      g1[2] = (int)(((td0 >> 16) & 0xFFFFu) | ((td1 & 0xFFFFu) << 16));
      g1[3] = (int)(((td1 >> 16) & 0xFFFFu) | (16u << 16));  // tile_dim0 = 16
      g1[4] = (int)(td1 & 0xFFFFu);                // tile_dim1 = kc, tile_dim2=0
      g1[5] = (int)(st0 & 0xFFFFFFFFu);            // dim0_stride lo
      g1[6] = (int)((st0 >> 32) & 0xFFFFu);        // dim0_stride hi
      g1[7] = 0;
      const v4i z4 = {};
#if defined(TDM_6ARG)
      const v8i z8 = {};
      __builtin_amdgcn_tensor_load_to_lds(g0, g1, z4, z4, z8, 0);
#else
      __builtin_amdgcn_tensor_load_to_lds(g0, g1, z4, z4, 0);
#endif
      __builtin_amdgcn_s_wait_tensorcnt(0);
    }
#else
    // ---- fallback: cooperative load Wp[k][c] = Wm[(kb+k)*N + n0 + c] ----
    for (int i = tid; i < (kc << 4); i += 128) {
      const int k = i >> 4, c = i & 15;
      Wp[(k << 4) + c] = Wm[(size_t)(kb + k) * N + n0 + c];
    }
#endif
    __syncthreads();

    const float* __restrict__ a_ptr = Arow + kb;
    if (kb + KCHUNK < K) __builtin_prefetch(a_ptr + KCHUNK, 0, 0);
#pragma unroll 4
    for (int kk = 0; kk < kc; kk += 4) {
      const v2f a = *(const v2f*)(a_ptr + kk);          // global_load_b64
      v2f b;
      b.x = Bcol[(kk + 2 * khalf) << 4];                // ds_load_b32 pair
      b.y = Bcol[(kk + 2 * khalf + 1) << 4];
      acc = __builtin_amdgcn_wmma_f32_16x16x4_f32(
          /*neg_a=*/false, a, /*neg_b=*/false, b,
          /*c_mod=*/(short)0, acc, /*reuse_a=*/false, /*reuse_b=*/false);
    }
    __syncthreads();
  }

  const float bi = bias ? bias[col] : 0.0f;
#pragma unroll
  for (int r = 0; r < 8; ++r) {
    const int rrow = m0 + r + 8 * khalf;
    if (rrow < M) C[(size_t)rrow * N + col] = acc[r] + bi;
  }
}

// ---------------------------------------------------------------------------
// Fused input layer: y = ReLU(LN(x @ W[3x64] + b) * g + be), wave per node.
// ---------------------------------------------------------------------------
__global__ void lin3_ln_relu(const float* __restrict__ X, const float* __restrict__ Wm,
                             const float* __restrict__ b, const float* __restrict__ g,
                             const float* __restrict__ be, float* __restrict__ Y, int M) {
  const int wave = (int)((blockIdx.x * blockDim.x + threadIdx.x) >> 5);
  const int lane = threadIdx.x & 31;
  if (wave >= M) return;
  const float x0 = X[(size_t)wave * 3 + 0];
  const float x1 = X[(size_t)wave * 3 + 1];
  const float x2 = X[(size_t)wave * 3 + 2];
  const int c0 = lane, c1 = lane + 32;
  float ya = x0 * Wm[c0] + x1 * Wm[64 + c0] + x2 * Wm[128 + c0] + b[c0];
  float yb = x0 * Wm[c1] + x1 * Wm[64 + c1] + x2 * Wm[128 + c1] + b[c1];
  float s = ya + yb;
#pragma unroll
  for (int off = 16; off; off >>= 1) s += __shfl_xor(s, off, 32);
  const float mean = s * (1.0f / 64.0f);
  const float da = ya - mean, db = yb - mean;
  float vs = da * da + db * db;
#pragma unroll
  for (int off = 16; off; off >>= 1) vs += __shfl_xor(vs, off, 32);
  const float inv = rsqrtf(vs * (1.0f / 64.0f) + 1e-5f);
  float* y = Y + (size_t)wave * 64;
  y[c0] = fmaxf(da * inv * g[c0] + be[c0], 0.0f);
  y[c1] = fmaxf(db * inv * g[c1] + be[c1], 0.0f);
}

// ---------------------------------------------------------------------------
// LayerNorm(width 64) + ReLU, one wave per row, 2 features per lane.
// ---------------------------------------------------------------------------
__global__ void ln_relu64(const float* __restrict__ X, const float* __restrict__ g,
                          const float* __restrict__ be, float* __restrict__ Y, int M) {
  const int wave = (int)((blockIdx.x * blockDim.x + threadIdx.x) >> 5);
  const int lane = threadIdx.x & 31;
  if (wave >= M) return;
  const float* x = X + (size_t)wave * 64;
  float a = x[lane], b = x[lane + 32];
  float s = a + b;
#pragma unroll
  for (int off = 16; off; off >>= 1) s += __shfl_xor(s, off, 32);
  const float mean = s * (1.0f / 64.0f);
  const float da = a - mean, db = b - mean;
  float vs = da * da + db * db;
#pragma unroll
  for (int off = 16; off; off >>= 1) vs += __shfl_xor(vs, off, 32);
  const float inv = rsqrtf(vs * (1.0f / 64.0f) + 1e-5f);
  float* y = Y + (size_t)wave * 64;
  y[lane]      = fmaxf(da * inv * g[lane]      + be[lane],      0.0f);
  y[lane + 32] = fmaxf(db * inv * g[lane + 32] + be[lane + 32], 0.0f);
}

// ---------------------------------------------------------------------------
// Edge pass 1: alpha[e,h] = scale * dot(q[dst,h,:], k[src,h,:]); segment max.
// ---------------------------------------------------------------------------
__global__ void edge_alpha(const int* __restrict__ ei, const float* __restrict__ Q,
                           const float* __restrict__ Kf, float* __restrict__ alpha,
                           unsigned* __restrict__ amax, int E) {
  const int t = blockIdx.x * blockDim.x + threadIdx.x;
  if (t >= E * HEADS) return;
  const int e = t >> 2, h = t & 3;
  const int src = ei[e], dst = ei[E + e];
  const float4* q = (const float4*)(Q  + (size_t)dst * HC + h * HD);
  const float4* k = (const float4*)(Kf + (size_t)src * HC + h * HD);
  float s = 0.0f;
#pragma unroll
  for (int i = 0; i < HD / 4; ++i) {
    const float4 a = q[i], b = k[i];
    s += a.x * b.x + a.y * b.y + a.z * b.z + a.w * b.w;
  }
  s *= 0.125f;  // 1/sqrt(64)
  alpha[t] = s;
  atomicMax(&amax[dst * HEADS + h], fenc(s));
}

// ---------------------------------------------------------------------------
// Edge pass 2: ex = exp(alpha - amax[dst]); segment sum into den.
// ---------------------------------------------------------------------------
__global__ void edge_exp(const int* __restrict__ ei, const float* __restrict__ alpha,
                         const unsigned* __restrict__ amax, float* __restrict__ ex,
                         float* __restrict__ den, int E) {
  const int t = blockIdx.x * blockDim.x + threadIdx.x;
  if (t >= E * HEADS) return;
  const int e = t >> 2, h = t & 3;
  const int dst = ei[E + e];
  const float m = fdec(amax[dst * HEADS + h]);
  const float v = __expf(alpha[t] - m);
  ex[t] = v;
  atomicAdd(&den[dst * HEADS + h], v);
}

// ---------------------------------------------------------------------------
// Edge pass 3: out[dst] += (ex/den) * v[src]; out pre-loaded with skip proj.
// ---------------------------------------------------------------------------
__global__ void edge_scatter(const int* __restrict__ ei, const float* __restrict__ ex,
                             const float* __restrict__ den, const float* __restrict__ V,
                             float* __restrict__ out, int E) {
  const long long t = (long long)blockIdx.x * blockDim.x + threadIdx.x;
  if (t >= (long long)E * (HC / 4)) return;
  const int e = (int)(t >> 6), c4 = (int)(t & 63);
  const int h = c4 >> 4;
  const int src = ei[e], dst = ei[E + e];
  const float w = ex[e * HEADS + h] / (den[dst * HEADS + h] + 1e-16f);
  const float4 v = ((const float4*)(V + (size_t)src * HC))[c4];
  float* o = out + (size_t)dst * HC + c4 * 4;
  atomicAdd(o + 0, v.x * w);
  atomicAdd(o + 1, v.y * w);
  atomicAdd(o + 2, v.z * w);
  atomicAdd(o + 3, v.w * w);
}

// ---------------------------------------------------------------------------
// Graph aggregation scatter: sum / sumsq / min / max, one thread per (node,f).
// ---------------------------------------------------------------------------
__global__ void agg_scatter(const float* __restrict__ H, const int* __restrict__ batch,
                            float* __restrict__ sum, float* __restrict__ sumsq,
                            unsigned* __restrict__ mn, unsigned* __restrict__ mx, int Nn) {
  const long long t = (long long)blockIdx.x * blockDim.x + threadIdx.x;
  if (t >= (long long)Nn * HC) return;
  const int n = (int)(t >> 8), f = (int)(t & 255);
  const int g = batch[n];
  const float v = H[t];
  const size_t o = (size_t)g * HC + f;
  atomicAdd(&sum[o], v);
  atomicAdd(&sumsq[o], v * v);
  atomicMin(&mn[o], fenc(v));
  atomicMax(&mx[o], fenc(v));
}

__global__ void count_nodes(const int* __restrict__ batch, float* __restrict__ cnt, int Nn) {
  const int n = blockIdx.x * blockDim.x + threadIdx.x;
  if (n < Nn) atomicAdd(&cnt[batch[n]], 1.0f);
}

// ---------------------------------------------------------------------------
// Finalize [G, 5*HC] = [sum, mean, min, max, std] concat.
// ---------------------------------------------------------------------------
__global__ void agg_finalize(const float* __restrict__ sum, const float* __restrict__ sumsq,
                             const unsigned* __restrict__ mn, const unsigned* __restrict__ mx,
                             const float* __restrict__ cnt, float* __restrict__ agg, int G) {
  const int t = blockIdx.x * blockDim.x + threadIdx.x;
  if (t >= G * HC) return;
  const int g = t >> 8, f = t & 255;
  const float c = cnt[g], cc = fmaxf(c, 1.0f);
  const float s = sum[t];
  const float mean = s / cc;
  const float mean2 = sumsq[t] / cc;
  const float var = mean2 - mean * mean;
  const float sd = sqrtf(fmaxf(var, 0.0f) + 1e-5f);
  const float mnv = (c > 0.0f) ? fdec(mn[t]) : 0.0f;
  const float mxv = (c > 0.0f) ? fdec(mx[t]) : 0.0f;
  float* o = agg + (size_t)g * (5 * HC);
  o[f]          = s;
  o[HC + f]     = mean;
  o[2 * HC + f] = mnv;
  o[3 * HC + f] = mxv;
  o[4 * HC + f] = sd;
}

// ---------------------------------------------------------------------------
// Final linear 64 -> 1: one wave per graph.
// ---------------------------------------------------------------------------
__global__ void final_linear(const float* __restrict__ X, const float* __restrict__ W,
                             const float* __restrict__ b, float* __restrict__ out, int G) {
  const int wave = (int)((blockIdx.x * blockDim.x + threadIdx.x) >> 5);
  const int lane = threadIdx.x & 31;
  if (wave >= G) return;
  const float* x = X + (size_t)wave * 64;
  float s = x[lane] * W[lane] + x[lane + 32] * W[lane + 32];
#pragma unroll
  for (int off = 16; off; off >>= 1) s += __shfl_xor(s, off, 32);
  if (lane == 0) out[wave] = s + b[0];
}

// ---------------------------------------------------------------------------
// Host orchestration.
// ---------------------------------------------------------------------------
static inline void launch_gemm(const float* A, const float* W, const float* bias,
                               float* C, int M, int K, int N, hipStream_t s) {
  dim3 grid((M + 63) / 64, N / 16);
  gemm_wmma_f32<<<grid, 128, 0, s>>>(A, W, bias, C, M, K, N);
}

static inline void launch_ln(const float* X, const float* g, const float* be,
                             float* Y, int M, hipStream_t s) {
  ln_relu64<<<(M + 7) / 8, 256, 0, s>>>(X, g, be, Y, M);
}

extern "C" void kernel_launch(void* const* d_in, const int* in_sizes, int n_in,
                              void* d_out, int out_size, void* d_ws, size_t ws_size,
                              hipStream_t stream) {
  // ---- inputs (setup_inputs dict insertion order, params tree flattened) ----
  const float* x   = (const float*)d_in[0];  // [N,3]
  const int* batch = (const int*)d_in[1];    // [N]
  const int* ei    = (const int*)d_in[2];    // [2,E]
  // d_in[3] = num_graphs (device scalar, unused; G = out_size)
  const int N = in_sizes[0] / 3;
  const int E = in_sizes[2] / 2;
  const int G = out_size;

  int p = 4;
  const float *iW[3], *ib[3], *ig[3], *ibe[3];
  for (int l = 0; l < 3; ++l) {
    iW[l] = (const float*)d_in[p++]; ib[l] = (const float*)d_in[p++];
    ig[l] = (const float*)d_in[p++]; ibe[l] = (const float*)d_in[p++];
  }
  const float *qW[3], *qb[3], *kW[3], *kb[3], *vW[3], *vb[3], *sW[3], *sb[3];
  for (int l = 0; l < 3; ++l) {
    qW[l] = (const float*)d_in[p++]; qb[l] = (const float*)d_in[p++];
    kW[l] = (const float*)d_in[p++]; kb[l] = (const float*)d_in[p++];
    vW[l] = (const float*)d_in[p++]; vb[l] = (const float*)d_in[p++];
    sW[l] = (const float*)d_in[p++]; sb[l] = (const float*)d_in[p++];
  }
  const float *rW[3], *rb[3], *rg[3], *rbe[3];
  for (int l = 0; l < 3; ++l) {
    rW[l] = (const float*)d_in[p++]; rb[l] = (const float*)d_in[p++];
    rg[l] = (const float*)d_in[p++]; rbe[l] = (const float*)d_in[p++];
  }
  const float* fW = (const float*)d_in[p++];
  const float* fb = (const float*)d_in[p++];
  (void)n_in; (void)ws_size;

  // ---- workspace bump allocator ----
  char* w = (char*)d_ws;
  auto alloc = [&](size_t bytes) -> void* {
    void* ptr = (void*)w;
    w += (bytes + 255) & ~(size_t)255;
    return ptr;
  };
  float* t0     = (float*)alloc((size_t)N * 64 * 4);
  float* t1     = (float*)alloc((size_t)N * 64 * 4);
  float* hA     = (float*)alloc((size_t)N * HC * 4);
  float* hB     = (float*)alloc((size_t)N * HC * 4);
  float* Qb     = (float*)alloc((size_t)N * HC * 4);
  float* Kb     = (float*)alloc((size_t)N * HC * 4);
  float* Vb     = (float*)alloc((size_t)N * HC * 4);
  float* alphaB = (float*)alloc((size_t)E * HEADS * 4);
  float* exB    = (float*)alloc((size_t)E * HEADS * 4);
  unsigned* amx = (unsigned*)alloc((size_t)N * HEADS * 4);
  float* den    = (float*)alloc((size_t)N * HEADS * 4);
  float* gsum   = (float*)alloc((size_t)G * HC * 4);
  float* gsq    = (float*)alloc((size_t)G * HC * 4);
  unsigned* gmn = (unsigned*)alloc((size_t)G * HC * 4);
  unsigned* gmx = (unsigned*)alloc((size_t)G * HC * 4);
  float* gcnt   = (float*)alloc((size_t)G * 4);
  float* agg    = (float*)alloc((size_t)G * 5 * HC * 4);
  float* r1     = (float*)alloc((size_t)G * 64 * 4);
  float* r2     = (float*)alloc((size_t)G * 64 * 4);

  // ---- input MLP: fused(3->64,LN,ReLU), then 2x [GEMM64 + LN+ReLU] ----
  lin3_ln_relu<<<(N + 7) / 8, 256, 0, stream>>>(x, iW[0], ib[0], ig[0], ibe[0], t1, N);
  launch_gemm(t1, iW[1], ib[1], t0, N, 64, 64, stream);
  launch_ln(t0, ig[1], ibe[1], t1, N, stream);
  launch_gemm(t1, iW[2], ib[2], t0, N, 64, 64, stream);
  launch_ln(t0, ig[2], ibe[2], t1, N, stream);

  // ---- TransformerConv layers ----
  const float* cur = t1;
  int curK = 64;
  float* cur_out = hA;
  float* nxt_out = hB;
  for (int l = 0; l < 3; ++l) {
    launch_gemm(cur, qW[l], qb[l], Qb, N, curK, HC, stream);
    launch_gemm(cur, kW[l], kb[l], Kb, N, curK, HC, stream);
    launch_gemm(cur, vW[l], vb[l], Vb, N, curK, HC, stream);
    launch_gemm(cur, sW[l], sb[l], cur_out, N, curK, HC, stream);  // skip proj

    hipMemsetAsync(amx, 0x00, (size_t)N * HEADS * 4, stream);  // enc(-inf) = 0
    hipMemsetAsync(den, 0x00, (size_t)N * HEADS * 4, stream);

    const int eh = E * HEADS;
    edge_alpha<<<(eh + 255) / 256, 256, 0, stream>>>(ei, Qb, Kb, alphaB, amx, E);
    edge_exp<<<(eh + 255) / 256, 256, 0, stream>>>(ei, alphaB, amx, exB, den, E);
    const long long esc = (long long)E * (HC / 4);
    edge_scatter<<<(unsigned)((esc + 255) / 256), 256, 0, stream>>>(ei, exB, den, Vb,
                                                                    cur_out, E);
    cur = cur_out;
    curK = HC;
    float* tmp = cur_out; cur_out = nxt_out; nxt_out = tmp;
  }

  // ---- multi-aggregation over graphs ----
  hipMemsetAsync(gsum, 0x00, (size_t)G * HC * 4, stream);
  hipMemsetAsync(gsq, 0x00, (size_t)G * HC * 4, stream);
  hipMemsetAsync(gmn, 0xFF, (size_t)G * HC * 4, stream);  // enc(+inf)
  hipMemsetAsync(gmx, 0x00, (size_t)G * HC * 4, stream);  // enc(-inf)
  hipMemsetAsync(gcnt, 0x00, (size_t)G * 4, stream);

  const long long nf = (long long)N * HC;
  agg_scatter<<<(unsigned)((nf + 255) / 256), 256, 0, stream>>>(cur, batch, gsum, gsq,
                                                                gmn, gmx, N);
  count_nodes<<<(N + 255) / 256, 256, 0, stream>>>(batch, gcnt, N);
  agg_finalize<<<(G * HC + 255) / 256, 256, 0, stream>>>(gsum, gsq, gmn, gmx, gcnt,
                                                         agg, G);

  // ---- regression MLP: 1280 -> 64 -> 64 -> 64 -> 1 ----
  launch_gemm(agg, rW[0], rb[0], r1, G, 5 * HC, 64, stream);
  launch_ln(r1, rg[0], rbe[0], r2, G, stream);
  launch_gemm(r2, rW[1], rb[1], r1, G, 64, 64, stream);
  launch_ln(r1, rg[1], rbe[1], r2, G, stream);
  launch_gemm(r2, rW[2], rb[2], r1, G, 64, 64, stream);
  launch_ln(r1, rg[2], rbe[2], r2, G, stream);

  final_linear<<<(G * 32 + 255) / 256, 256, 0, stream>>>(r2, fW, fb, (float*)d_out, G);
}